// PSMCosineModule_60567628808605
// MI455X (gfx1250) — compile-verified
//
#include <hip/hip_runtime.h>

typedef __attribute__((ext_vector_type(2))) float v2f;
typedef __attribute__((ext_vector_type(8))) float v8f;

#define B_      4
#define C_      512
#define H_      96
#define W_      320
#define DEPTH_  48
#define KC      32            // channels staged per LDS chunk
#define NCHUNK  (C_ / KC)     // 16 chunks
#define LPAD    20            // padded LDS row (dwords) for 16-wide left rows
#define RPAD    68            // padded LDS row (dwords) for 64-wide right rows

// GLOBAL_LOAD_ASYNC_TO_LDS_B128, GVS addressing (ISA §10.2 / §15.18):
//   mem_addr = SGPR64 base + VGPR32 byte offset;  16B/lane -> LDS, ASYNCcnt.
// Per-thread voffset is loop-invariant; the uniform base advances in SALU.
__device__ __forceinline__ void async_cp16(unsigned lds_off, unsigned goff,
                                           const float* base) {
    asm volatile("global_load_async_to_lds_b128 %0, %1, %2"
                 :: "v"(lds_off), "v"(goff), "s"(base)
                 : "memory");
}

__device__ __forceinline__ unsigned lds_addr(const float* l) {
    return (unsigned)(size_t)(__attribute__((address_space(3))) float*)l;
}

__device__ __forceinline__ void wait_async0() {
#if __has_builtin(__builtin_amdgcn_s_wait_asynccnt)
    __builtin_amdgcn_s_wait_asynccnt(0);
#else
    asm volatile("s_wait_asynccnt 0" ::: "memory");
#endif
}

// One block per (b, h, w-tile); 4 waves. Wave d computes the 16x16 output tile
// at diagonal offset i = 16d + m - n, contracting C=512 with
// v_wmma_f32_16x16x4_f32 (exact fp32). LDS is double buffered; chunk c+1
// streams in via the async Global->LDS engine while chunk c feeds the WMMAs.
__global__ __launch_bounds__(128) void cost_volume_wmma(
    const float* __restrict__ left,
    const float* __restrict__ right,
    float* __restrict__ out)
{
    __shared__ __align__(16) float Ls[2][KC][LPAD];  // left  tile: KC x 16 used
    __shared__ __align__(16) float Rs[2][KC][RPAD];  // right strip: KC x 64 used

    const int w0   = blockIdx.x * 16;
    const int h    = blockIdx.y;
    const int b    = blockIdx.z;
    const int tid  = threadIdx.x;
    const int lane = tid & 31;
    const int wave = tid >> 5;           // diagonal index d
    const int n    = lane & 15;
    const int hi   = lane >> 4;

    const size_t plane = (size_t)H_ * W_;
    const size_t base  = (size_t)b * C_ * plane + (size_t)h * W_;
    const float* Lc = left  + base + w0;        // advances by KC*plane per chunk
    const float* Rc = right + base + (w0 - 48); // advances by KC*plane per chunk

    const bool active = (w0 >= 16 * wave);
    const int  off    = 48 - 16 * wave;             // B-tile column offset in Rs
    const int  jmiss  = (w0 < 48) ? (48 - w0) : 0;  // w' < 0 zero band (mult of 16)

    // Pre-zero the w' < 0 band in both buffers; loads never touch it.
    if (jmiss > 0 && tid < KC) {
        const float4 z = make_float4(0.f, 0.f, 0.f, 0.f);
        for (int j = 0; j < jmiss; j += 4) {
            *(float4*)&Rs[0][tid][j] = z;
            *(float4*)&Rs[1][tid][j] = z;
        }
    }

    // ---- loop-invariant per-thread addresses ----
    const int lk = tid >> 2;                     // left: one float4 per thread
    const int lj = (tid & 3) << 2;
    const unsigned goffL = (unsigned)(((size_t)lk * plane + lj) * sizeof(float));
    unsigned ldsL[2] = { lds_addr(&Ls[0][lk][lj]), lds_addr(&Ls[1][lk][lj]) };

    unsigned goffR[4], ldsR[2][4];
    int      rpred[4];
    #pragma unroll
    for (int q = 0; q < 4; ++q) {
        const int idx = tid + q * 128;
        const int rk  = idx >> 4;
        const int rj  = (idx & 15) << 2;
        goffR[q]   = (unsigned)(((size_t)rk * plane + rj) * sizeof(float));
        ldsR[0][q] = lds_addr(&Rs[0][rk][rj]);
        ldsR[1][q] = lds_addr(&Rs[1][rk][rj]);
        rpred[q]   = (rj >= jmiss);
    }

    v8f acc = {};

    auto load_chunk = [&](int bufsel) {
        async_cp16(ldsL[bufsel], goffL, Lc);
        #pragma unroll
        for (int q = 0; q < 4; ++q)
            if (rpred[q])                // EXEC-masked async VMEM is legal
                async_cp16(ldsR[bufsel][q], goffR[q], Rc);
        Lc += KC * plane;
        Rc += KC * plane;
    };

    load_chunk(0);

    #pragma unroll 2
    for (int c = 0; c < NCHUNK; ++c) {
        wait_async0();                   // chunk c fully landed in LDS
        __syncthreads();                 // visible to all waves; prior reads drained
        if (c + 1 < NCHUNK)
            load_chunk((c + 1) & 1);     // overlaps compute below

        if (active) {
            const int cur = c & 1;
            #pragma unroll
            for (int k4 = 0; k4 < KC / 4; ++k4) {
                // A 16x4 f32: lanes 0-15 hold K,K+1; lanes 16-31 hold K+2,K+3
                const int kb = k4 * 4 + (hi << 1);
                v2f a, bb;
                a[0]  = Ls[cur][kb    ][n];
                a[1]  = Ls[cur][kb + 1][n];
                bb[0] = Rs[cur][kb    ][off + n];
                bb[1] = Rs[cur][kb + 1][off + n];
                acc = __builtin_amdgcn_wmma_f32_16x16x4_f32(
                    false, a, false, bb, (short)0, acc, false, false);
            }
        }
    }

    // D layout: VGPR v -> M = v + 8*hi, N = lane&15. i = 16d + m - n.
    if (active) {
        const float scale = 1.0f / (float)C_;
        const int   HW    = H_ * W_;
        float* outp = out + ((size_t)b * DEPTH_ * H_ + h) * W_ + w0;  // + i*HW + m
        const int ibase = 16 * wave - n;
        #pragma unroll
        for (int v = 0; v < 8; ++v) {
            const int m = v + 8 * hi;
            const int i = ibase + m;
            if ((unsigned)i < (unsigned)DEPTH_)
                outp[(size_t)i * HW + m] = acc[v] * scale;
        }
    }
}

extern "C" void kernel_launch(void* const* d_in, const int* in_sizes, int n_in,
                              void* d_out, int out_size, void* d_ws, size_t ws_size,
                              hipStream_t stream) {
    const float* left  = (const float*)d_in[0];
    const float* right = (const float*)d_in[1];
    float* out = (float*)d_out;

    // Zero the w < i region (never written by any tile); graph-capture safe.
    (void)hipMemsetAsync(out, 0, (size_t)out_size * sizeof(float), stream);

    dim3 grid(W_ / 16, H_, B_);   // 20 x 96 x 4 = 7680 blocks
    cost_volume_wmma<<<grid, 128, 0, stream>>>(left, right, out);
}